// TripletCenterLoss_90623809945949
// MI455X (gfx1250) — compile-verified
//
#include <hip/hip_runtime.h>
#include <hip/hip_bf16.h>

typedef __attribute__((ext_vector_type(2))) float v2f;
typedef __attribute__((ext_vector_type(8))) float v8f;

#define B_SIZE   8192
#define C_CLASS  100
#define D_DIM    128
#define NT       7          // 7 N-tiles of 16 -> 112 padded classes
#define CPAD     112
#define LSTRIDE  132        // LDS floats per center row (128 + 4 pad, kills bank conflicts)
#define SSTRIDE  116        // LDS floats per S-tile row
#define MARGIN   5.0f

// ---------------- prep: class-presence bitmap + center squared norms ----------------
__global__ __launch_bounds__(128) void tcl_prep(const int* __restrict__ targets,
                                                const float* __restrict__ centers,
                                                unsigned* __restrict__ bitmap,
                                                float* __restrict__ c2) {
    __shared__ unsigned sbm[4];
    int tid = threadIdx.x;
    if (tid < 4) sbm[tid] = 0u;
    __syncthreads();
    for (int i = tid; i < B_SIZE; i += 128) {
        int t = targets[i];
        atomicOr(&sbm[t >> 5], 1u << (t & 31));
    }
    __syncthreads();
    if (tid < 4) bitmap[tid] = sbm[tid];
    if (tid < C_CLASS) {
        const float* cr = centers + tid * D_DIM;
        float s = 0.f;
        #pragma unroll 4
        for (int d = 0; d < D_DIM; d += 4) {
            float4 v = *(const float4*)(cr + d);
            s += v.x * v.x + v.y * v.y + v.z * v.z + v.w * v.w;
        }
        c2[tid] = s;
    }
}

// ---------------- main: WMMA GEMM tile (16 rows x 112 classes) + per-row hinge ----------------
__global__ __launch_bounds__(32) void tcl_main(const float* __restrict__ inputs,
                                               const int* __restrict__ targets,
                                               const float* __restrict__ centers,
                                               const unsigned* __restrict__ bitmap,
                                               const float* __restrict__ c2,
                                               float* __restrict__ row_loss) {
    __shared__ float lds[CPAD * LSTRIDE];   // 59136 B; reused as S-tile afterwards
    const int lane = threadIdx.x;           // wave32: one wave per block
    const int m0   = blockIdx.x * 16;

    // Stage centers into LDS: row c at stride LSTRIDE; pad rows 100..111 with zeros.
    for (int c = 0; c < CPAD; ++c) {
        float4 v;
        if (c < C_CLASS) v = ((const float4*)(centers + c * D_DIM))[lane];
        else             v = make_float4(0.f, 0.f, 0.f, 0.f);
        *(float4*)(&lds[c * LSTRIDE + lane * 4]) = v;
    }
    __syncthreads();

    // WMMA fragment addressing per CDNA5 ISA layout:
    //   A 16x4 f32: lanes 0-15 hold (K0,K1) of row lane; lanes 16-31 hold (K2,K3).
    //   B 4x16 f32: lanes 0-15 hold (K0,K1) of col lane; lanes 16-31 hold (K2,K3).
    const int   kg   = (lane >> 4) * 2;                 // 0 or 2
    const int   mrow = lane & 15;
    const float* arow = inputs + (size_t)(m0 + mrow) * D_DIM;

    v8f acc[NT] = {};
    for (int k0 = 0; k0 < D_DIM; k0 += 4) {
        v2f a = *(const v2f*)(arow + k0 + kg);
        #pragma unroll
        for (int t = 0; t < NT; ++t) {
            v2f b = *(const v2f*)(&lds[(t * 16 + mrow) * LSTRIDE + k0 + kg]);
            acc[t] = __builtin_amdgcn_wmma_f32_16x16x4_f32(
                false, a, false, b, (short)0, acc[t], false, false);
        }
    }
    __syncthreads();

    // Spill D tiles to LDS (reuse buffer): D VGPR r = row r (lanes 0-15) / row r+8 (lanes 16-31).
    #pragma unroll
    for (int t = 0; t < NT; ++t) {
        #pragma unroll
        for (int r = 0; r < 8; ++r) {
            int mm = r + ((lane >> 4) << 3);
            int nn = mrow + t * 16;
            lds[mm * SSTRIDE + nn] = acc[t][r];
        }
    }
    __syncthreads();

    if (lane < 16) {
        const int row = m0 + lane;
        // x2 = ||inputs[row]||^2
        const float* xr = inputs + (size_t)row * D_DIM;
        float x2 = 0.f;
        #pragma unroll 4
        for (int d = 0; d < D_DIM; d += 4) {
            float4 v = *(const float4*)(xr + d);
            x2 += v.x * v.x + v.y * v.y + v.z * v.z + v.w * v.w;
        }
        const int tgt = targets[row];
        const float ap_d2 = x2 + c2[tgt] - 2.f * lds[lane * SSTRIDE + tgt];
        const float ap    = sqrtf(fmaxf(ap_d2, 1e-12f));

        // hardest negative: min d2 over present classes != tgt (sqrt/clamp are monotone)
        float mind2 = 3.402823466e38f;
        for (int c = 0; c < C_CLASS; ++c) {
            bool present = (bitmap[c >> 5] >> (c & 31)) & 1u;
            if (present && c != tgt) {
                float d2 = x2 + c2[c] - 2.f * lds[lane * SSTRIDE + c];
                mind2 = fminf(mind2, d2);
            }
        }
        float loss;
        if (mind2 > 3.3e38f) {
            loss = 0.f;                     // no negative class present -> relu(-inf) = 0
        } else {
            float an = sqrtf(fmaxf(mind2, 1e-12f));
            loss = fmaxf(0.f, MARGIN + ap - an);
        }
        row_loss[row] = loss;
    }
}

// ---------------- deterministic mean reduction ----------------
__global__ __launch_bounds__(256) void tcl_reduce(const float* __restrict__ row_loss,
                                                  float* __restrict__ out) {
    __shared__ float red[256];
    int tid = threadIdx.x;
    float s = 0.f;
    for (int i = tid; i < B_SIZE; i += 256) s += row_loss[i];
    red[tid] = s;
    __syncthreads();
    for (int off = 128; off > 0; off >>= 1) {
        if (tid < off) red[tid] += red[tid + off];
        __syncthreads();
    }
    if (tid == 0) out[0] = red[0] * (1.0f / (float)B_SIZE);
}

extern "C" void kernel_launch(void* const* d_in, const int* in_sizes, int n_in,
                              void* d_out, int out_size, void* d_ws, size_t ws_size,
                              hipStream_t stream) {
    const float* inputs  = (const float*)d_in[0];
    const int*   targets = (const int*)d_in[1];
    const float* centers = (const float*)d_in[2];
    float*       out     = (float*)d_out;

    // workspace layout: [0,16) bitmap (4 u32) | [32,432) c2 (100 f32) | [512, 512+32768) row_loss
    unsigned* bitmap   = (unsigned*)d_ws;
    float*    c2       = (float*)d_ws + 8;
    float*    row_loss = (float*)d_ws + 128;

    tcl_prep  <<<1,               128, 0, stream>>>(targets, centers, bitmap, c2);
    tcl_main  <<<B_SIZE / 16,      32, 0, stream>>>(inputs, targets, centers, bitmap, c2, row_loss);
    tcl_reduce<<<1,               256, 0, stream>>>(row_loss, out);
}